// GNNEncoder_28518582845817
// MI455X (gfx1250) — compile-verified
//
#include <hip/hip_runtime.h>
#include <hip/hip_bf16.h>
#include <float.h>

// ---------------------------------------------------------------------------
// GAT encoder for MI455X (gfx1250). Dense GEMMs use V_WMMA_F32_16X16X4_F32
// (fp32 A/B, fp32 accum). Edge aggregation uses native f32 global atomics.
// ---------------------------------------------------------------------------

typedef float v2f __attribute__((ext_vector_type(2)));
typedef float v8f __attribute__((ext_vector_type(8)));

#define HDIM 64
#define OUTD 128
#define GNUM 1000
#define EDIM 3

static inline int cdivl(long a, int b) { return (int)((a + b - 1) / b); }

// ------------------------------ utility ------------------------------------
__global__ void fill_kernel(float* __restrict__ p, float v, long n) {
  long i = (long)blockIdx.x * blockDim.x + threadIdx.x;
  if (i < n) p[i] = v;
}

// ------------------------------ WMMA GEMM ----------------------------------
// C[M,Nc] = A[M,K] @ W[K,Nc] (+ bias).  M, K, Nc multiples of 16/4.
// One wave computes one 16x16 tile of C, accumulating K in steps of 4 with
// V_WMMA_F32_16X16X4_F32.
// A fragment (16x4 f32): lanes 0-15 hold K = k,k+1 ; lanes 16-31 K = k+2,k+3.
// B fragment (4x16 f32): same K split, N = lane&15.
// C (16x16 f32, 8 VGPRs): VGPR j = row j (lanes 0-15) / row j+8 (lanes 16-31).
__global__ void wmma_gemm_f32(const float* __restrict__ A,
                              const float* __restrict__ W,
                              const float* __restrict__ bias,
                              float* __restrict__ C,
                              int M, int K, int Nc) {
  int lane = threadIdx.x & 31;
  int wid  = blockIdx.x * (blockDim.x >> 5) + (threadIdx.x >> 5);
  int mtiles = M >> 4, ntiles = Nc >> 4;
  if (wid >= mtiles * ntiles) return;   // whole-wave uniform exit
  int tm = wid % mtiles, tn = wid / mtiles;
  int half = lane >> 4;                 // 0: K lo pair, 1: K hi pair
  int l15  = lane & 15;
  int arow = (tm << 4) + l15;
  int col  = (tn << 4) + l15;
  v8f acc = {0.f, 0.f, 0.f, 0.f, 0.f, 0.f, 0.f, 0.f};
  for (int k = 0; k < K; k += 4) {
    int ka = k + 2 * half;
    v2f a, b;
    a.x = A[(long)arow * K + ka];
    a.y = A[(long)arow * K + ka + 1];
    b.x = W[(long)ka * Nc + col];
    b.y = W[(long)(ka + 1) * Nc + col];
    acc = __builtin_amdgcn_wmma_f32_16x16x4_f32(false, a, false, b,
                                                (short)0, acc, false, false);
  }
  float bv = bias ? bias[col] : 0.f;
#pragma unroll
  for (int j = 0; j < 8; ++j) {
    int r = (tm << 4) + j + 8 * half;
    C[(long)r * Nc + col] = acc[j] + bv;
  }
}

// --------------------- self-loop attr (scatter mean) -----------------------
__global__ void loop_accum_kernel(const long long* __restrict__ ei,
                                  const float* __restrict__ eattr,
                                  float* __restrict__ deg,
                                  float* __restrict__ lsum, int E) {
  int e = blockIdx.x * blockDim.x + threadIdx.x;
  if (e >= E) return;
  int d = (int)ei[E + e];
  atomicAdd(&deg[d], 1.f);
#pragma unroll
  for (int j = 0; j < EDIM; ++j)
    atomicAdd(&lsum[(long)d * EDIM + j], eattr[(long)e * EDIM + j]);
}

__global__ void loop_final_kernel(const float* __restrict__ deg,
                                  const float* __restrict__ lsum,
                                  float* __restrict__ lattr, int N) {
  int i = blockIdx.x * blockDim.x + threadIdx.x;
  if (i >= N) return;
  float dv = fmaxf(deg[i], 1.f);
#pragma unroll
  for (int j = 0; j < EDIM; ++j)
    lattr[(long)i * EDIM + j] = lsum[(long)i * EDIM + j] / dv;
}

// --------------------- per-layer small precomputes -------------------------
// we_att[d] = sum_h W_e[d,h] * att_edge[h]   (collapses edge-score GEMV)
__global__ void weatt_kernel(const float* __restrict__ We,
                             const float* __restrict__ att_e,
                             float* __restrict__ weatt) {
  int d = threadIdx.x;
  if (d < EDIM) {
    float s = 0.f;
    for (int h = 0; h < HDIM; ++h) s += We[d * HDIM + h] * att_e[h];
    weatt[d] = s;
  }
}

__global__ void scores_kernel(const float* __restrict__ hlin,
                              const float* __restrict__ att_s,
                              const float* __restrict__ att_d,
                              float* __restrict__ a_src,
                              float* __restrict__ a_dst, int N) {
  int i = blockIdx.x * blockDim.x + threadIdx.x;
  if (i >= N) return;
  float ss = 0.f, sd = 0.f;
  for (int f = 0; f < HDIM; ++f) {
    float v = hlin[(long)i * HDIM + f];
    ss += v * att_s[f];
    sd += v * att_d[f];
  }
  a_src[i] = ss;
  a_dst[i] = sd;
}

// --------------------------- edge softmax ----------------------------------
__device__ inline void atomicMaxF(float* addr, float v) {
  if (v >= 0.f) atomicMax((int*)addr, __float_as_int(v));
  else          atomicMin((unsigned int*)addr, __float_as_uint(v));
}

__global__ void alpha_kernel(const long long* __restrict__ ei,
                             const float* __restrict__ eattr,
                             const float* __restrict__ lattr,
                             const float* __restrict__ weatt,
                             const float* __restrict__ a_src,
                             const float* __restrict__ a_dst,
                             float* __restrict__ alpha,
                             float* __restrict__ amax, int E, int N) {
  int t = blockIdx.x * blockDim.x + threadIdx.x;
  if (t >= E + N) return;
  int s, d;
  float ae;
  if (t < E) {
    s = (int)ei[t]; d = (int)ei[E + t];
    const float* ea = &eattr[(long)t * EDIM];
    ae = ea[0] * weatt[0] + ea[1] * weatt[1] + ea[2] * weatt[2];
  } else {
    s = d = t - E;
    const float* la = &lattr[(long)s * EDIM];
    ae = la[0] * weatt[0] + la[1] * weatt[1] + la[2] * weatt[2];
  }
  float a = a_src[s] + a_dst[d] + ae;
  a = a > 0.f ? a : 0.2f * a;            // leaky_relu(0.2)
  alpha[t] = a;
  atomicMaxF(&amax[d], a);
}

__global__ void exp_kernel(const long long* __restrict__ ei,
                           float* __restrict__ alpha,
                           const float* __restrict__ amax,
                           float* __restrict__ denom, int E, int N) {
  int t = blockIdx.x * blockDim.x + threadIdx.x;
  if (t >= E + N) return;
  int d = (t < E) ? (int)ei[E + t] : (t - E);
  float ex = expf(alpha[t] - amax[d]);
  alpha[t] = ex;                          // reuse as ex
  atomicAdd(&denom[d], ex);
}

// one thread per (edge, feature): out[dst] += w * h[src]
__global__ void scatter_kernel(const long long* __restrict__ ei,
                               const float* __restrict__ ex,
                               const float* __restrict__ denom,
                               const float* __restrict__ h,
                               float* __restrict__ out, int E, int N) {
  long tid = (long)blockIdx.x * blockDim.x + threadIdx.x;
  long t = tid >> 6;
  int f = (int)(tid & 63);
  if (t >= (long)E + N) return;
  int s, d;
  if (t < E) { s = (int)ei[t]; d = (int)ei[E + t]; }
  else       { s = d = (int)(t - E); }
  float w = ex[t] / (denom[d] + 1e-16f);
  atomicAdd(&out[(long)d * HDIM + f], w * h[(long)s * HDIM + f]);
}

// ------------------ bias + leaky(0.01) + optional layernorm ----------------
__global__ void post_kernel(const float* __restrict__ acc,
                            const float* __restrict__ bias,
                            const float* __restrict__ g,
                            const float* __restrict__ bln,
                            float* __restrict__ out, int do_ln) {
  __shared__ float s[HDIM];
  __shared__ float mu, var;
  int i = blockIdx.x;
  int f = threadIdx.x;                    // 64 threads
  float v = acc[(long)i * HDIM + f] + bias[f];
  v = v > 0.f ? v : 0.01f * v;
  if (do_ln) {
    s[f] = v; __syncthreads();
    for (int st = 32; st > 0; st >>= 1) { if (f < st) s[f] += s[f + st]; __syncthreads(); }
    if (f == 0) mu = s[0] * (1.f / HDIM);
    __syncthreads();
    float ddv = v - mu;
    s[f] = ddv * ddv; __syncthreads();
    for (int st = 32; st > 0; st >>= 1) { if (f < st) s[f] += s[f + st]; __syncthreads(); }
    if (f == 0) var = s[0] * (1.f / HDIM);
    __syncthreads();
    v = (v - mu) * rsqrtf(var + 1e-5f) * g[f] + bln[f];
  }
  out[(long)i * HDIM + f] = v;
}

// ------------------------------ pooling ------------------------------------
__global__ void count_kernel(const long long* __restrict__ batch,
                             float* __restrict__ counts, int N) {
  int i = blockIdx.x * blockDim.x + threadIdx.x;
  if (i >= N) return;
  atomicAdd(&counts[(int)batch[i]], 1.f);
}

__global__ void pool_accum_kernel(const long long* __restrict__ batch,
                                  const float* __restrict__ h,
                                  float* __restrict__ out, int N) {
  long tid = (long)blockIdx.x * blockDim.x + threadIdx.x;
  long i = tid >> 7;
  int f = (int)(tid & 127);
  if (i >= N) return;
  atomicAdd(&out[(long)batch[i] * OUTD + f], h[i * OUTD + f]);
}

__global__ void pool_div_kernel(float* __restrict__ out,
                                const float* __restrict__ counts, int G) {
  int t = blockIdx.x * blockDim.x + threadIdx.x;
  if (t >= G * OUTD) return;
  out[t] /= fmaxf(counts[t / OUTD], 1.f);
}

// ------------------------------ host side ----------------------------------
static void launch_gemm(const float* A, const float* W, const float* bias,
                        float* C, int M, int K, int Nc, hipStream_t s) {
  int tiles = (M / 16) * (Nc / 16);
  int blocks = (tiles + 3) / 4;           // 4 waves / 128-thread block
  wmma_gemm_f32<<<blocks, 128, 0, s>>>(A, W, bias, C, M, K, Nc);
}

extern "C" void kernel_launch(void* const* d_in, const int* in_sizes, int n_in,
                              void* d_out, int out_size, void* d_ws, size_t ws_size,
                              hipStream_t stream) {
  // ---- input mapping (setup_inputs insertion order; params leaves in order)
  const float*     x     = (const float*)d_in[0];         // [N, F_IN]
  const long long* ei    = (const long long*)d_in[1];     // [2, E] int64
  const float*     eattr = (const float*)d_in[2];         // [E, 3]
  const long long* batch = (const long long*)d_in[3];     // [N] int64
  // convs[i]: W, att_src, att_dst, W_e, att_edge, b  -> indices 4+6i .. 9+6i
  // lns: g0(22), b0(23), g1(24), b1(25); mp_W1(26), mp_b1(27), mp_W2(28), mp_b2(29)
  const int N = in_sizes[3];
  const int E = in_sizes[2] / EDIM;
  const int F_IN = in_sizes[0] / N;
  const int T = E + N;                    // edges incl. self loops

  float* out = (float*)d_out;             // [G, OUTD]

  // ---- workspace layout (floats)
  float* ws = (float*)d_ws;
  long off = 0;
  float* lin   = ws + off; off += (long)N * HDIM;
  float* acc   = ws + off; off += (long)N * HDIM;
  float* cur   = ws + off; off += (long)N * HDIM;
  float* hout  = ws + off; off += (long)N * OUTD;
  float* alpha = ws + off; off += (long)T;
  float* a_src = ws + off; off += N;
  float* a_dst = ws + off; off += N;
  float* amax  = ws + off; off += N;
  float* denom = ws + off; off += N;
  float* deg   = ws + off; off += N;
  float* lsum  = ws + off; off += (long)N * EDIM;
  float* lattr = ws + off; off += (long)N * EDIM;
  float* weatt = ws + off; off += 64;
  float* cnts  = ws + off; off += GNUM;

  const int TB = 256;

  // ---- self-loop edge attributes (scatter mean of eattr by dst), once
  fill_kernel<<<cdivl((long)N * (1 + EDIM), TB), TB, 0, stream>>>(deg, 0.f, (long)N * (1 + EDIM)); // deg+lsum contiguous
  loop_accum_kernel<<<cdivl(E, TB), TB, 0, stream>>>(ei, eattr, deg, lsum, E);
  loop_final_kernel<<<cdivl(N, TB), TB, 0, stream>>>(deg, lsum, lattr, N);

  // ---- three GAT conv layers
  for (int i = 0; i < 3; ++i) {
    const float* W     = (const float*)d_in[4 + 6 * i];
    const float* att_s = (const float*)d_in[5 + 6 * i];
    const float* att_d = (const float*)d_in[6 + 6 * i];
    const float* W_e   = (const float*)d_in[7 + 6 * i];
    const float* att_e = (const float*)d_in[8 + 6 * i];
    const float* bconv = (const float*)d_in[9 + 6 * i];
    const float* hin   = (i == 0) ? x : cur;
    const int    K     = (i == 0) ? F_IN : HDIM;

    launch_gemm(hin, W, nullptr, lin, N, K, HDIM, stream);     // h = xin @ W
    weatt_kernel<<<1, 32, 0, stream>>>(W_e, att_e, weatt);
    scores_kernel<<<cdivl(N, TB), TB, 0, stream>>>(lin, att_s, att_d, a_src, a_dst, N);

    fill_kernel<<<cdivl(N, TB), TB, 0, stream>>>(amax, -FLT_MAX, N);
    fill_kernel<<<cdivl(N, TB), TB, 0, stream>>>(denom, 0.f, N);
    fill_kernel<<<cdivl((long)N * HDIM, TB), TB, 0, stream>>>(acc, 0.f, (long)N * HDIM);

    alpha_kernel<<<cdivl(T, TB), TB, 0, stream>>>(ei, eattr, lattr, weatt,
                                                  a_src, a_dst, alpha, amax, E, N);
    exp_kernel<<<cdivl(T, TB), TB, 0, stream>>>(ei, alpha, amax, denom, E, N);
    scatter_kernel<<<cdivl((long)T * HDIM, TB), TB, 0, stream>>>(ei, alpha, denom,
                                                                 lin, acc, E, N);

    const float* g   = (i < 2) ? (const float*)d_in[22 + 2 * i] : nullptr;
    const float* bln = (i < 2) ? (const float*)d_in[23 + 2 * i] : nullptr;
    post_kernel<<<N, HDIM, 0, stream>>>(acc, bconv,
                                        g ? g : bconv, bln ? bln : bconv,
                                        cur, (i < 2) ? 1 : 0);
  }

  // ---- MLP head (WMMA GEMMs with fused bias)
  const float* mpW1 = (const float*)d_in[26];
  const float* mpb1 = (const float*)d_in[27];
  const float* mpW2 = (const float*)d_in[28];
  const float* mpb2 = (const float*)d_in[29];
  launch_gemm(cur, mpW1, mpb1, lin,  N, HDIM, HDIM, stream);   // [N,64]
  launch_gemm(lin, mpW2, mpb2, hout, N, HDIM, OUTD, stream);   // [N,128]

  // ---- mean pool by graph
  fill_kernel<<<cdivl(GNUM, TB), TB, 0, stream>>>(cnts, 0.f, GNUM);
  fill_kernel<<<cdivl((long)GNUM * OUTD, TB), TB, 0, stream>>>(out, 0.f, (long)GNUM * OUTD);
  count_kernel<<<cdivl(N, TB), TB, 0, stream>>>(batch, cnts, N);
  pool_accum_kernel<<<cdivl((long)N * OUTD, TB), TB, 0, stream>>>(batch, hout, out, N);
  pool_div_kernel<<<cdivl((long)GNUM * OUTD, TB), TB, 0, stream>>>(out, cnts, GNUM);
}